// SoftplusNormCausal_7602092114034
// MI455X (gfx1250) — compile-verified
//
#include <hip/hip_runtime.h>

// SoftplusNormCausal for MI455X (gfx1250).
// Memory-bound row-normalize: stage causal prefix of each row into LDS via
// CDNA5 async global->LDS copies (ASYNCcnt), one HBM read pass, NT stores.

#define SEQ   2048
#define TPB   256          // 8 wave32 per block
#define EPSF  1e-12f
#define ALPHAF 2.0f
#define THETAF 0.5f

typedef __attribute__((ext_vector_type(4))) float v4f;

__global__ __launch_bounds__(TPB) void spnorm_causal_kernel(
    const float* __restrict__ x, float* __restrict__ out) {
  __shared__ float srow[SEQ];          // 8 KB staged row
  __shared__ float partial[TPB / 32];  // per-wave partial sums
  __shared__ float total_s;

  const int row_flat = blockIdx.x;          // b*S + i, rows are contiguous
  const int i = row_flat & (SEQ - 1);       // causal row index within matrix
  const long long base = (long long)row_flat * SEQ;
  const float* xrow = x + base;             // uniform -> SGPR pair
  float* orow = out + base;
  const int t = threadIdx.x;

  // 32-bit LDS byte address of srow: low 32 bits of the flat shared pointer
  // are the LDS offset (shared aperture encodes base in the upper bits).
  unsigned lds_base = (unsigned)(unsigned long long)(void*)srow;

  // ---- async stage: 512 x 16B chunks; chunk c covers elems [4c,4c+3].
  // Only chunks intersecting the causal prefix (4c <= i) are fetched.
  {
    const unsigned c0 = (unsigned)t;
    const unsigned c1 = (unsigned)t + TPB;
    if ((int)(4u * c0) <= i) {
      unsigned lo = lds_base + c0 * 16u;
      unsigned go = c0 * 16u;
      asm volatile("global_load_async_to_lds_b128 %0, %1, %2"
                   :: "v"(lo), "v"(go), "s"(xrow) : "memory");
    }
    if ((int)(4u * c1) <= i) {
      unsigned lo = lds_base + c1 * 16u;
      unsigned go = c1 * 16u;
      asm volatile("global_load_async_to_lds_b128 %0, %1, %2"
                   :: "v"(lo), "v"(go), "s"(xrow) : "memory");
    }
  }
  asm volatile("s_wait_asynccnt 0x0" ::: "memory");  // this wave's copies landed
  __syncthreads();                                   // all waves' copies visible

  // ---- compute y = squareplus(2(x-0.5))^2 on causal prefix, accumulate sum
  v4f y0 = {}, y1 = {};
  float lsum = 0.0f;
  const int j0 = 4 * t;
  const int j1 = 4 * t + (SEQ / 2);

  if (j0 <= i) {
    v4f xa = *(const v4f*)&srow[j0];
#pragma unroll
    for (int k = 0; k < 4; ++k) {
      if (j0 + k <= i) {
        float arg = ALPHAF * (xa[k] - THETAF);
        float sp = 0.5f * (arg + __builtin_sqrtf(arg * arg + 4.0f));
        float yv = sp * sp;
        y0[k] = yv;
        lsum += yv;
      }
    }
  }
  if (j1 <= i) {
    v4f xb = *(const v4f*)&srow[j1];
#pragma unroll
    for (int k = 0; k < 4; ++k) {
      if (j1 + k <= i) {
        float arg = ALPHAF * (xb[k] - THETAF);
        float sp = 0.5f * (arg + __builtin_sqrtf(arg * arg + 4.0f));
        float yv = sp * sp;
        y1[k] = yv;
        lsum += yv;
      }
    }
  }

  // ---- row-sum reduction: shfl-xor within wave32, then tiny LDS tree
#pragma unroll
  for (int off = 16; off > 0; off >>= 1)
    lsum += __shfl_xor(lsum, off, 32);

  const int wave = t >> 5;
  if ((t & 31) == 0) partial[wave] = lsum;
  __syncthreads();
  if (t == 0) {
    float s = EPSF;
#pragma unroll
    for (int w = 0; w < TPB / 32; ++w) s += partial[w];
    total_s = s;
  }
  __syncthreads();

  const float inv = 1.0f / total_s;
  y0 *= inv;
  y1 *= inv;

  // ---- NT b128 stores: result is write-once, keep it out of L2
  __builtin_nontemporal_store(y0, (v4f*)&orow[j0]);
  __builtin_nontemporal_store(y1, (v4f*)&orow[j1]);
}

extern "C" void kernel_launch(void* const* d_in, const int* in_sizes, int n_in,
                              void* d_out, int out_size, void* d_ws, size_t ws_size,
                              hipStream_t stream) {
  (void)n_in; (void)out_size; (void)d_ws; (void)ws_size;
  const float* x = (const float*)d_in[0];
  float* out = (float*)d_out;
  const int rows = in_sizes[0] / SEQ;  // 2*8*2048 = 32768 rows
  spnorm_causal_kernel<<<rows, TPB, 0, stream>>>(x, out);
}